// VectorQuantize_4157528343151
// MI455X (gfx1250) — compile-verified
//
#include <hip/hip_runtime.h>
#include <hip/hip_bf16.h>
#include <math.h>

typedef __attribute__((ext_vector_type(2))) float v2f;
typedef __attribute__((ext_vector_type(8))) float v8f;
typedef __attribute__((ext_vector_type(4))) int   v4i;

#define Bn 16
#define Dn 512
#define Tn 4096
#define CBn 1024
#define CDn 8
#define TT 128           // tokens per block
#define NTHREADS 256     // 8 waves of 32

// workspace layout (float offsets)
#define WS_WIN   0                 // 8*512   weight-normed in_proj
#define WS_WOUT  4096              // 512*8   weight-normed out_proj
#define WS_CBN   8192              // 1024*8  normalized codebook
#define WS_CBSQ  16384             // 1024    ||cb_n||^2
#define WS_LOSS  17408             // 16      per-batch loss accumulators

#define WMMA_F32X4(A, B, C) \
  __builtin_amdgcn_wmma_f32_16x16x4_f32(false, (A), false, (B), 0, (C), false, false)

#if __has_builtin(__builtin_amdgcn_global_load_async_to_lds_b128)
#define HAVE_ASYNC_LDS 1
typedef __attribute__((address_space(1))) v4i g_v4i;   // global int4
typedef __attribute__((address_space(3))) v4i l_v4i;   // LDS int4
static __device__ __forceinline__ void wait_async0() {
#if __has_builtin(__builtin_amdgcn_s_wait_asynccnt)
  __builtin_amdgcn_s_wait_asynccnt(0);
#else
  asm volatile("s_wait_asynccnt 0x0" ::: "memory");
#endif
}
#else
#define HAVE_ASYNC_LDS 0
#endif

// ---------------------------------------------------------------------------
// Kernel 1: weight-norm + codebook normalization + loss accumulator reset
// ---------------------------------------------------------------------------
__global__ void vq_prep(const float* v_in, const float* g_in,
                        const float* v_out, const float* g_out,
                        const float* codebook, float* ws) {
  __shared__ float s_scale[CDn];
  const int tid = threadIdx.x;
  if (tid < 16) ws[WS_LOSS + tid] = 0.0f;
  if (tid < CDn) {
    float ss = 0.0f;
    for (int i = 0; i < Dn; ++i) { float v = v_in[tid * Dn + i]; ss += v * v; }
    s_scale[tid] = g_in[tid] / sqrtf(ss);
  }
  __syncthreads();
  for (int j = tid; j < CDn * Dn; j += NTHREADS) {
    int o = j / Dn;
    ws[WS_WIN + j] = v_in[j] * s_scale[o];
  }
  for (int r = tid; r < Dn; r += NTHREADS) {
    float ss = 0.0f;
    for (int c = 0; c < CDn; ++c) { float v = v_out[r * CDn + c]; ss += v * v; }
    float sc = g_out[r] / sqrtf(ss);
    for (int c = 0; c < CDn; ++c) ws[WS_WOUT + r * CDn + c] = v_out[r * CDn + c] * sc;
  }
  for (int r = tid; r < CBn; r += NTHREADS) {
    float ss = 0.0f;
    for (int c = 0; c < CDn; ++c) { float v = codebook[r * CDn + c]; ss += v * v; }
    float inv = 1.0f / fmaxf(sqrtf(ss), 1e-12f);
    float sq = 0.0f;
    for (int c = 0; c < CDn; ++c) {
      float vn = codebook[r * CDn + c] * inv;
      ws[WS_CBN + r * CDn + c] = vn;
      sq += vn * vn;
    }
    ws[WS_CBSQ + r] = sq;
  }
}

// ---------------------------------------------------------------------------
// Kernel 2: fused in_proj (WMMA) -> nearest-codebook (WMMA) -> gather ->
//           losses -> out_proj (WMMA). One block = 128 tokens of one batch.
// ---------------------------------------------------------------------------
__global__ __launch_bounds__(NTHREADS) void vq_main(
    const float* __restrict__ z, const float* __restrict__ b_in,
    const float* __restrict__ b_out, const float* __restrict__ codebook,
    const float* ws, float* ws_loss,
    float* __restrict__ out, float* __restrict__ out_idx) {
  __shared__ __align__(16) float s_cb[CBn * CDn];   // 32 KB normalized codebook
  __shared__ __align__(16) float s_win[CDn * Dn];   // 16 KB w_in
  __shared__ float s_cbsq[CBn];                     //  4 KB
  __shared__ float s_enc[8 * 16 * CDn];             //  4 KB normalized encodings
  __shared__ float s_zq[TT * CDn];                  //  4 KB gathered entries
  __shared__ int   s_idx[TT];
  __shared__ float s_bout[Dn];                      //  2 KB
  __shared__ float s_bin[CDn];
  __shared__ float s_red[8];

  const int tid   = threadIdx.x;
  const int lane  = tid & 31;
  const int wv    = tid >> 5;      // wave 0..7
  const int half  = lane >> 4;     // 0: lanes 0-15, 1: lanes 16-31
  const int l15   = lane & 15;
  const int b     = blockIdx.y;
  const int tile0 = blockIdx.x * TT;
  const int tok0  = tile0 + wv * 16;   // this wave's 16 tokens

  // ---- stage block-resident constants into LDS ----
#if HAVE_ASYNC_LDS
  // 32 KB codebook + 16 KB w_in via async DMA to LDS (ASYNCcnt path)
  for (int j = tid * 4; j < CBn * CDn; j += NTHREADS * 4)
    __builtin_amdgcn_global_load_async_to_lds_b128(
        (g_v4i*)(ws + WS_CBN + j), (l_v4i*)&s_cb[j], 0, 0);
  for (int j = tid * 4; j < CDn * Dn; j += NTHREADS * 4)
    __builtin_amdgcn_global_load_async_to_lds_b128(
        (g_v4i*)(ws + WS_WIN + j), (l_v4i*)&s_win[j], 0, 0);
#else
  for (int j = tid; j < CBn * CDn; j += NTHREADS) s_cb[j] = ws[WS_CBN + j];
  for (int j = tid; j < CDn * Dn; j += NTHREADS) s_win[j] = ws[WS_WIN + j];
#endif
  for (int j = tid; j < CBn; j += NTHREADS) s_cbsq[j] = ws[WS_CBSQ + j];
  for (int j = tid; j < Dn; j += NTHREADS) s_bout[j] = b_out[j];
  if (tid < CDn) s_bin[tid] = b_in[tid];
#if HAVE_ASYNC_LDS
  wait_async0();
#endif
  __syncthreads();

  // ---- Phase I: z_e = w_in(8x512) @ z(512 x 16tok), M padded 8->16 ----
  // A layout (16x4 f32): lanes0-15 M=lane K={0,1}; lanes16-31 M=lane-16 K={2,3}
  // B layout (4x16 f32): lanes0-15 K={0,1} N=lane; lanes16-31 K={2,3}
  const float* zb = z + (size_t)b * Dn * Tn;
  v8f c = {0.f, 0.f, 0.f, 0.f, 0.f, 0.f, 0.f, 0.f};
  const int mm = l15 & 7;
  const bool mvalid = (l15 < 8);
  for (int k = 0; k < Dn; k += 4) {
    if (k + 8 < Dn)
      __builtin_prefetch(&zb[(size_t)(k + 8 + 2 * half) * Tn + tok0 + l15], 0, 3);
    v2f a, bb;
    float a0 = s_win[mm * Dn + k + 2 * half];
    float a1 = s_win[mm * Dn + k + 2 * half + 1];
    a.x = mvalid ? a0 : 0.0f;
    a.y = mvalid ? a1 : 0.0f;
    const int d0 = k + 2 * half;
    bb.x = __builtin_nontemporal_load(&zb[(size_t)d0 * Tn + tok0 + l15]);
    bb.y = __builtin_nontemporal_load(&zb[(size_t)(d0 + 1) * Tn + tok0 + l15]);
    c = WMMA_F32X4(a, bb, c);
  }
  // C layout: c[r] lanes0-15 = z_e[ch r][tok0+lane]; lanes16-31 = rows 8..15 = 0
  float ze[CDn];
  float ss = 0.0f;
#pragma unroll
  for (int r = 0; r < CDn; ++r) {
    ze[r] = c[r] + s_bin[r];
    ss += ze[r] * ze[r];
  }
  const float scale = 1.0f / fmaxf(sqrtf(ss), 1e-12f);
  if (lane < 16) {
#pragma unroll
    for (int r = 0; r < CDn; ++r)
      s_enc[(wv * 16 + l15) * CDn + r] = ze[r] * scale;
  }
  __syncthreads();

  // ---- Phase II: dist = ||cb_n||^2 - 2 * enc_n . cb_n via WMMA, argmin ----
  const int eb = (wv * 16 + l15) * CDn;   // A: M=token=l15 (both halves)
  v2f a0d, a1d;
  a0d.x = s_enc[eb + 2 * half];     a0d.y = s_enc[eb + 2 * half + 1];
  a1d.x = s_enc[eb + 4 + 2 * half]; a1d.y = s_enc[eb + 4 + 2 * half + 1];

  float bd[CDn];
  int   bi[CDn];
#pragma unroll
  for (int r = 0; r < CDn; ++r) { bd[r] = 3.402823e38f; bi[r] = CBn; }

#pragma unroll 2
  for (int e0 = 0; e0 < CBn; e0 += 16) {
    v2f b0, b1;
    const int cbb = (e0 + l15) * CDn;   // B[k][n] = cb_n[e0+n][k]
    b0.x = s_cb[cbb + 2 * half];     b0.y = s_cb[cbb + 2 * half + 1];
    b1.x = s_cb[cbb + 4 + 2 * half]; b1.y = s_cb[cbb + 4 + 2 * half + 1];
    v8f dacc = {0.f, 0.f, 0.f, 0.f, 0.f, 0.f, 0.f, 0.f};
    dacc = WMMA_F32X4(a0d, b0, dacc);
    dacc = WMMA_F32X4(a1d, b1, dacc);
    const float csq = s_cbsq[e0 + l15];
#pragma unroll
    for (int r = 0; r < CDn; ++r) {
      float dist = csq - 2.0f * dacc[r];   // token = r + 8*half, entry = e0+l15
      if (dist < bd[r]) { bd[r] = dist; bi[r] = e0 + l15; }
    }
  }
  // butterfly argmin inside each 16-lane half (halves hold different tokens)
#pragma unroll
  for (int m = 1; m <= 8; m <<= 1) {
#pragma unroll
    for (int r = 0; r < CDn; ++r) {
      float od = __shfl_xor(bd[r], m, 32);
      int   oi = __shfl_xor(bi[r], m, 32);
      if (od < bd[r] || (od == bd[r] && oi < bi[r])) { bd[r] = od; bi[r] = oi; }
    }
  }
  if (l15 == 0) {
#pragma unroll
    for (int r = 0; r < CDn; ++r) s_idx[wv * 16 + half * 8 + r] = bi[r];
  }
  __syncthreads();

  // ---- gather UNnormalized codebook rows + emit indices ----
  for (int j = tid; j < TT * CDn; j += NTHREADS) {
    int tk = j >> 3, ch = j & 7;
    s_zq[j] = codebook[s_idx[tk] * CDn + ch];
  }
  if (tid < TT) out_idx[b * Tn + tile0 + tid] = (float)s_idx[tid];
  __syncthreads();

  // ---- losses: mean (z_e - z_q)^2 (commitment == codebook in fwd) ----
  float ls = 0.0f;
#pragma unroll
  for (int r = 0; r < CDn; ++r) {
    float d = ze[r] - s_zq[(wv * 16 + l15) * CDn + r];
    ls += d * d;
  }
  ls = (lane < 16) ? ls : 0.0f;
#pragma unroll
  for (int m = 1; m < 32; m <<= 1) ls += __shfl_xor(ls, m, 32);
  if (lane == 0) s_red[wv] = ls;
  __syncthreads();
  if (tid == 0) {
    float tot = 0.0f;
    for (int w = 0; w < 8; ++w) tot += s_red[w];
    atomicAdd(&ws_loss[b], tot);
  }

  // ---- Phase IV: out = w_out(512x8) @ z_q(8 x 16tok) + b_out via WMMA ----
  const float* wo = ws + WS_WOUT;
  const int tb = (wv * 16 + l15) * CDn;  // B[k][n] = z_q[tok0+n][k]
  v2f b0o, b1o;
  b0o.x = s_zq[tb + 2 * half];     b0o.y = s_zq[tb + 2 * half + 1];
  b1o.x = s_zq[tb + 4 + 2 * half]; b1o.y = s_zq[tb + 4 + 2 * half + 1];
  float* outb = out + (size_t)b * Dn * Tn;
  for (int m0 = 0; m0 < Dn; m0 += 16) {
    const int row = m0 + l15;
    v2f a0o, a1o;
    a0o.x = wo[row * CDn + 2 * half];     a0o.y = wo[row * CDn + 2 * half + 1];
    a1o.x = wo[row * CDn + 4 + 2 * half]; a1o.y = wo[row * CDn + 4 + 2 * half + 1];
    v8f o8 = {0.f, 0.f, 0.f, 0.f, 0.f, 0.f, 0.f, 0.f};
    o8 = WMMA_F32X4(a0o, b0o, o8);
    o8 = WMMA_F32X4(a1o, b1o, o8);
#pragma unroll
    for (int r = 0; r < CDn; ++r) {
      const int orow = m0 + r + 8 * half;
      __builtin_nontemporal_store(o8[r] + s_bout[orow],
                                  &outb[(size_t)orow * Tn + tok0 + l15]);
    }
  }
}

// ---------------------------------------------------------------------------
// Kernel 3: finalize losses
// ---------------------------------------------------------------------------
__global__ void vq_finalize(const float* ws, float* out_loss) {
  const int tid = threadIdx.x;
  if (tid < 16) {
    float v = ws[WS_LOSS + tid] * (1.0f / ((float)CDn * (float)Tn));
    out_loss[tid] = v;        // commitment_loss
    out_loss[16 + tid] = v;   // codebook_loss (same forward value)
  }
}

extern "C" void kernel_launch(void* const* d_in, const int* in_sizes, int n_in,
                              void* d_out, int out_size, void* d_ws, size_t ws_size,
                              hipStream_t stream) {
  (void)in_sizes; (void)n_in; (void)out_size; (void)ws_size;
  const float* z     = (const float*)d_in[0];
  const float* v_in  = (const float*)d_in[1];
  const float* g_in  = (const float*)d_in[2];
  const float* b_in  = (const float*)d_in[3];
  const float* v_out = (const float*)d_in[4];
  const float* g_out = (const float*)d_in[5];
  const float* b_out = (const float*)d_in[6];
  const float* cb    = (const float*)d_in[7];
  float* ws  = (float*)d_ws;
  float* out = (float*)d_out;
  float* out_loss = out + (size_t)Bn * Dn * Tn;   // [16]+[16]
  float* out_idx  = out_loss + 32;                // [B*T] as float

  vq_prep<<<1, NTHREADS, 0, stream>>>(v_in, g_in, v_out, g_out, cb, ws);
  dim3 grid(Tn / TT, Bn);
  vq_main<<<grid, NTHREADS, 0, stream>>>(z, b_in, b_out, cb, ws, ws + WS_LOSS,
                                         out, out_idx);
  vq_finalize<<<1, 64, 0, stream>>>(ws, out_loss);
}